// MLP_DiffPhys_81415400063377
// MI455X (gfx1250) — compile-verified
//
#include <hip/hip_runtime.h>
#include <hip/hip_bf16.h>

typedef float v2f __attribute__((ext_vector_type(2)));
typedef float v8f __attribute__((ext_vector_type(8)));

#define HDIM 64
#define LDS_STRIDE 66   // padded stride (dwords): banks = 2*col + kr -> conflict-free b64

// gfx1250 hardware tanh (TRANS32 op, co-executes with VALU)
__device__ __forceinline__ float tanh_hw(float x) {
    float y;
    asm("v_tanh_f32 %0, %1" : "=v"(y) : "v"(x));
    return y;
}

__device__ __forceinline__ float softplusf(float x) {
    // jax.nn.softplus = log1p(exp(x)); guard overflow
    return (x > 20.0f) ? x : log1pf(__expf(x));
}

// One wave (32 lanes) computes beta(t)*dt for 16 consecutive time rows using
// V_WMMA_F32_16X16X4_F32 tiles for the 64x64 middle layer.
// W2 staged TRANSPOSED (+padded) in LDS so each lane's B-fragment (K=kr,kr+1
// at fixed column) is one aligned ds_load_b64 into a WMMA-ready register pair.
__global__ __launch_bounds__(256) void beta_mlp_kernel(
    const float* __restrict__ t,  const float* __restrict__ W1,
    const float* __restrict__ b1, const float* __restrict__ W2,
    const float* __restrict__ b2, const float* __restrict__ W3,
    const float* __restrict__ b3, float* __restrict__ out, int T)
{
    __shared__ float sW2t[HDIM * LDS_STRIDE];
    // stage W2 (64x64 row-major) transposed: sW2t[c*66 + r] = W2[r*64 + c]
    for (int i = threadIdx.x; i < HDIM * HDIM; i += blockDim.x) {
        const int r = i >> 6, c = i & 63;
        sW2t[c * LDS_STRIDE + r] = W2[i];
    }
    __syncthreads();

    const int lane = threadIdx.x & 31;
    const int half = lane >> 4;      // which 16-lane half
    const int mrow = lane & 15;      // A-matrix row / B-matrix column
    const int koff = half * 2;       // K offset within each K=4 chunk (ISA A layout)
    const int tile = blockIdx.x * (blockDim.x >> 5) + (threadIdx.x >> 5);
    const int row0 = tile * 16;
    if (row0 >= T) return;           // wave-uniform exit (EXEC stays all-ones for WMMA)

    const float dt = t[1] - t[0];
    int trow = row0 + mrow; if (trow > T - 1) trow = T - 1;
    const float tm = t[trow];

    // Layer 1: h1 = tanh(t*W1 + b1), produced directly as 16 A-fragments (16x4 f32)
    float a0[16], a1[16];
#pragma unroll
    for (int kc = 0; kc < 16; ++kc) {
        const int c = kc * 4 + koff;
        a0[kc] = tanh_hw(fmaf(tm, W1[c],     b1[c]));
        a1[kc] = tanh_hw(fmaf(tm, W1[c + 1], b1[c + 1]));
    }

    // Layer 2 via WMMA: 4 N-tiles x 16 K-chunks, fused with layer 3 partial dot
    float p[8] = {0.f,0.f,0.f,0.f,0.f,0.f,0.f,0.f};
    for (int nt = 0; nt < 4; ++nt) {
        const int col = nt * 16 + mrow;
        // column of W2 in LDS; (col*66 + kr) is even -> 8B-aligned v2f loads
        const v2f* bcol = (const v2f*)(&sW2t[col * LDS_STRIDE + koff]);
        v8f c = {};
#pragma unroll
        for (int kc = 0; kc < 16; ++kc) {
            v2f av; av.x = a0[kc]; av.y = a1[kc];
            const v2f bv = bcol[kc * 2];            // = {W2[kr][col], W2[kr+1][col]}
            c = __builtin_amdgcn_wmma_f32_16x16x4_f32(
                    false, av, false, bv, (short)0, c, false, false);
        }
        const float b2n = b2[col];
        const float w3n = W3[col];   // W3 is (64,1)
#pragma unroll
        for (int v = 0; v < 8; ++v) {
            const float h2 = tanh_hw(c[v] + b2n);   // row M = half*8+v, col = col
            p[v] = fmaf(h2, w3n, p[v]);
        }
    }

    // Reduce layer-3 dot across the 16 lanes of each half (rows 0-7 / 8-15)
#pragma unroll
    for (int m = 8; m >= 1; m >>= 1)
#pragma unroll
        for (int v = 0; v < 8; ++v)
            p[v] += __shfl_xor(p[v], m, 32);

    if (mrow == 0) {
        const float b3v = b3[0];
#pragma unroll
        for (int v = 0; v < 8; ++v) {
            const int r = half * 8 + v;
            if (row0 + r < T)
                out[row0 + r] = softplusf(p[v] + b3v) * dt;   // store beta*dt
        }
    }
}

// Inherently-serial SIR recurrence: single lane, chunked b128 loads of beta*dt
// (independent of the dependence chain, so they pipeline out of L2), then
// overwrite the same slots with I. d_out acts as its own scratch.
__global__ void sir_scan_kernel(const float* __restrict__ t,
                                const float* __restrict__ I_init,
                                const float* __restrict__ gamma_param,
                                float* __restrict__ out, int T)
{
    if (threadIdx.x != 0 || blockIdx.x != 0) return;
    const float dt    = t[1] - t[0];
    const float gamma = softplusf(gamma_param[0]);
    const float cI    = 1.0f - gamma * dt;      // I' = cI*I + (beta*dt)*S*I
    float I = I_init[0];
    float S = 1.0f - I;

    int base = 0;
    for (; base + 16 <= T; base += 16) {
        const float4* bp = (const float4*)(out + base);
        const float4 c0 = bp[0], c1 = bp[1], c2 = bp[2], c3 = bp[3];
        const float b[16] = {c0.x,c0.y,c0.z,c0.w, c1.x,c1.y,c1.z,c1.w,
                             c2.x,c2.y,c2.z,c2.w, c3.x,c3.y,c3.z,c3.w};
        float r[16];
#pragma unroll
        for (int j = 0; j < 16; ++j) {
            const float x  = S * I;
            const float bx = b[j] * x;
            S = fminf(fmaxf(S - bx, 0.0f), 1.0f);
            I = fminf(fmaxf(fmaf(cI, I, bx), 0.0f), 1.0f);
            r[j] = I;
        }
        float4* op = (float4*)(out + base);
        op[0] = make_float4(r[0], r[1], r[2],  r[3]);
        op[1] = make_float4(r[4], r[5], r[6],  r[7]);
        op[2] = make_float4(r[8], r[9], r[10], r[11]);
        op[3] = make_float4(r[12],r[13],r[14], r[15]);
    }
    for (; base < T; ++base) {
        const float bdt = out[base];
        const float x = S * I, bx = bdt * x;
        S = fminf(fmaxf(S - bx, 0.0f), 1.0f);
        I = fminf(fmaxf(fmaf(cI, I, bx), 0.0f), 1.0f);
        out[base] = I;
    }
}

extern "C" void kernel_launch(void* const* d_in, const int* in_sizes, int n_in,
                              void* d_out, int out_size, void* d_ws, size_t ws_size,
                              hipStream_t stream) {
    const float* t  = (const float*)d_in[0];
    const float* Ii = (const float*)d_in[1];
    const float* W1 = (const float*)d_in[2];
    const float* b1 = (const float*)d_in[3];
    const float* W2 = (const float*)d_in[4];
    const float* b2 = (const float*)d_in[5];
    const float* W3 = (const float*)d_in[6];
    const float* b3 = (const float*)d_in[7];
    const float* gp = (const float*)d_in[8];
    float* out = (float*)d_out;
    const int T = in_sizes[0];

    const int tiles = (T + 15) / 16;
    const int wavesPerBlock = 8;                     // 256 threads = 8 wave32
    const int grid = (tiles + wavesPerBlock - 1) / wavesPerBlock;
    beta_mlp_kernel<<<grid, 256, 0, stream>>>(t, W1, b1, W2, b2, W3, b3, out, T);
    sir_scan_kernel<<<1, 32, 0, stream>>>(t, Ii, gp, out, T);
}